// PosMLP_8546984919444
// MI455X (gfx1250) — compile-verified
//
#include <hip/hip_runtime.h>

typedef float v2f __attribute__((ext_vector_type(2)));
typedef float v8f __attribute__((ext_vector_type(8)));

// D = A(16x4) * B(4x16) + C, all f32, wave32. 8-arg form:
// (neg_a, A, neg_b, B, c_mod, C, reuse_a, reuse_b)
#define WMMA_F32_K4(A, B, C) \
  __builtin_amdgcn_wmma_f32_16x16x4_f32(false, (A), false, (B), (short)0, (C), false, false)

// ---------------------------------------------------------------------------
// Stage 1: weights[2400][65] = queries[2400][256] @ Wg[256][65] + bg[65]
// One wave per 16x16 output tile; K=256 accumulated in 64 WMMA steps.
// Tile grid: 150 M-tiles x 5 N-tiles (N padded 65 -> 80, guarded).
// ---------------------------------------------------------------------------
__global__ __launch_bounds__(32) void posmlp_weights_gemm(
    const float* __restrict__ q,   // 2400 x 256
    const float* __restrict__ Wg,  // 256 x 65
    const float* __restrict__ bg,  // 65
    float* __restrict__ wt)        // 2400 x 65 (workspace)
{
  const int t    = blockIdx.x;       // 0..749
  const int m0   = (t / 5) * 16;
  const int n0   = (t % 5) * 16;
  const int lane = threadIdx.x & 31;
  const int m    = lane & 15;
  const bool low = lane < 16;

  const int col   = n0 + m;
  const int ccol  = (col < 65) ? col : 64;         // clamp -> always in-bounds load
  const float cms = (col < 65) ? 1.0f : 0.0f;      // zero out padded columns

  const float* qrow = q + (size_t)(m0 + m) * 256;
  const int khalf = low ? 0 : 2;                   // A/B K-pair per lane half

  v8f acc = {};
  #pragma unroll 4
  for (int k0 = 0; k0 < 256; k0 += 4) {
    const int kk = k0 + khalf;
    v2f a;                                         // A: row m0+m, K = kk, kk+1
    a.x = qrow[kk];
    a.y = qrow[kk + 1];
    v2f b;                                         // B: col n0+m, K = kk, kk+1
    b.x = Wg[(size_t)kk * 65 + ccol] * cms;
    b.y = Wg[(size_t)(kk + 1) * 65 + ccol] * cms;
    acc = WMMA_F32_K4(a, b, acc);
  }

  const float bgv = bg[ccol];
  const int rbase = m0 + (low ? 0 : 8);            // D: VGPR v -> row v (+8 high half)
  #pragma unroll
  for (int v = 0; v < 8; ++v) {
    if (col < 65) wt[(size_t)(rbase + v) * 65 + col] = acc[v] + bgv;
  }
}

// ---------------------------------------------------------------------------
// Stage 2: per query n, render 64x64 map.
//   pre[k][pix] = xs*ax[k] + ys*ay[k] + 1*c[k]   (one WMMA f32 16x16x4)
//   out[pix]    = sum_k relu(pre)*w2[k] + b2     (8 max + 8 fma + shfl_xor(16))
// Block = 256 threads (8 waves) per query. Wave tile: 16 x-pixels, 32 y-rows,
// processed 2 rows/iteration (2 WMMAs) for a full 32-lane coalesced store.
// ---------------------------------------------------------------------------
__global__ __launch_bounds__(256) void posmlp_render(
    const float* __restrict__ pos,  // 2400 x 4
    const float* __restrict__ wt,   // 2400 x 65
    float* __restrict__ out)        // 2400 x 64 x 64
{
  const int n    = blockIdx.x;        // query index
  const int lane = threadIdx.x & 31;
  const int wave = threadIdx.x >> 5;
  const int m    = lane & 15;
  const bool low = lane < 16;

  const int x0    = (wave & 3) * 16;  // x stripe
  const int ybase = (wave >> 2) * 32; // y half

  const float4 p = ((const float4*)pos)[n];
  const float cx = p.x, cy = p.y, bw = p.z, bh = p.w;
  const float* w = wt + (size_t)n * 65;

  // Per-lane A-matrix row (hidden index m): [ax, ay, c, 0]
  const float ax = w[m]      / bw;
  const float ay = w[16 + m] / bh;
  const float cc = w[32 + m] - cx * ax - cy * ay;

  v2f A;
  A.x = low ? ax : cc;                // K=0 (low) / K=2 (high)
  A.y = low ? ay : 0.0f;              // K=1 (low) / K=3 (high)

  float w2h[8];                       // w2 slice matching this half's D rows
  const int w2off = 48 + (low ? 0 : 8);
  #pragma unroll
  for (int j = 0; j < 8; ++j) w2h[j] = w[w2off + j];
  const float b2 = w[64];

  const float xs = ((float)(x0 + m) + 0.5f) * (1.0f / 64.0f);
  const float bx = low ? xs : 1.0f;   // B K=0 row (low) / K=2 "1" row (high)

  float* orow = out + (size_t)n * 4096 + x0 + m;

  #pragma unroll 2
  for (int pr = 0; pr < 16; ++pr) {
    const int y0 = ybase + 2 * pr;
    const float ys0 = ((float)y0 + 0.5f) * (1.0f / 64.0f);
    const float ys1 = ((float)y0 + 1.5f) * (1.0f / 64.0f);

    v2f B0; B0.x = bx; B0.y = low ? ys0 : 0.0f;   // K=1 ys row / K=3 zero row
    v2f B1; B1.x = bx; B1.y = low ? ys1 : 0.0f;

    v8f d0 = {}; v8f d1 = {};
    d0 = WMMA_F32_K4(A, B0, d0);                  // pre for row y0, 16 pix x 16 hid
    d1 = WMMA_F32_K4(A, B1, d1);                  // pre for row y0+1

    float s0 = 0.0f, s1 = 0.0f;
    #pragma unroll
    for (int v = 0; v < 8; ++v) {                 // relu + dot with w2 (half each)
      s0 = fmaf(fmaxf(d0[v], 0.0f), w2h[v], s0);
      s1 = fmaf(fmaxf(d1[v], 0.0f), w2h[v], s1);
    }
    // combine hidden halves across lane groups (lane p <-> lane p+16)
    const float t0 = s0 + __shfl_xor(s0, 16, 32);
    const float t1 = s1 + __shfl_xor(s1, 16, 32);

    const float val = (low ? t0 : t1) + b2;       // low half stores row y0, high y0+1
    const int yoff = low ? 0 : 64;
    orow[(size_t)y0 * 64 + yoff] = val;
  }
}

// ---------------------------------------------------------------------------
extern "C" void kernel_launch(void* const* d_in, const int* in_sizes, int n_in,
                              void* d_out, int out_size, void* d_ws, size_t ws_size,
                              hipStream_t stream) {
  (void)in_sizes; (void)n_in; (void)out_size; (void)ws_size;
  const float* pos     = (const float*)d_in[0];  // 8 x 300 x 4
  const float* queries = (const float*)d_in[1];  // 8 x 300 x 256
  const float* Wg      = (const float*)d_in[2];  // 256 x 65
  const float* bg      = (const float*)d_in[3];  // 65
  // d_in[4] = size (== 64, fixed by problem shapes)

  float* wt = (float*)d_ws;                      // 2400 x 65 f32 = 624 KB scratch

  posmlp_weights_gemm<<<750, 32, 0, stream>>>(queries, Wg, bg, wt);
  posmlp_render<<<2400, 256, 0, stream>>>(pos, wt, (float*)d_out);
}